// IFNode_6708738916814
// MI455X (gfx1250) — compile-verified
//
#include <hip/hip_runtime.h>

// IF-neuron multi-step scan (hard reset), T x B x D fp32.
//   h_t = x_t + v_{t-1};  s_t = (h_t - 1.0 >= 0);  v_t = s_t ? 0 : h_t
// Output: spike sequence s (T,B,D) fp32.
//
// Memory-bound streaming kernel: one thread owns 4 consecutive D-elements
// and walks the T dimension with the membrane potential v held in VGPRs.
// 128-bit coalesced global ops with CDNA5 non-temporal (TH=NT) hints —
// 256 MB single-use working set should bypass, not thrash, the 192 MB L2.

typedef __attribute__((ext_vector_type(4))) float f4;

__global__ __launch_bounds__(256) void if_scan_v4(
    const float* __restrict__ x_seq,   // (T, N) flattened
    const float* __restrict__ v_init,  // (N)
    float* __restrict__ out,           // (T, N)
    int n4,                            // N / 4
    int T,
    int stride)                        // N elements per timestep
{
    const int i = blockIdx.x * blockDim.x + threadIdx.x;
    if (i >= n4) return;
    const int base = i * 4;

    f4 v = *(const f4*)(v_init + base);

    const float* xp = x_seq + base;
    float*       op = out   + base;

    // Loads are address-independent of the v-recurrence; unroll-by-4 keeps
    // several global_load_b128 in flight per wave (split LOADcnt waits).
#pragma unroll 4
    for (int t = 0; t < T; ++t) {
        f4 x = __builtin_nontemporal_load((const f4*)xp);
        f4 h = x + v;
        f4 s, vn;
#pragma unroll
        for (int k = 0; k < 4; ++k) {
            const bool fire = (h[k] - 1.0f) >= 0.0f;   // H(h - v_th), v_th = 1
            s[k]  = fire ? 1.0f : 0.0f;                // spike
            vn[k] = fire ? 0.0f : h[k];                // hard reset to v_reset = 0
        }
        __builtin_nontemporal_store(s, (f4*)op);
        v = vn;
        xp += stride;
        op += stride;
    }
}

// Scalar tail for N % 4 != 0 (not exercised for N = 1,048,576; kept for generality).
__global__ __launch_bounds__(256) void if_scan_tail(
    const float* __restrict__ x_seq,
    const float* __restrict__ v_init,
    float* __restrict__ out,
    int start, int n, int T, int stride)
{
    const int i = start + blockIdx.x * blockDim.x + threadIdx.x;
    if (i >= n) return;
    float v = v_init[i];
    for (int t = 0; t < T; ++t) {
        const float h = x_seq[(size_t)t * stride + i] + v;
        const bool fire = (h - 1.0f) >= 0.0f;
        out[(size_t)t * stride + i] = fire ? 1.0f : 0.0f;
        v = fire ? 0.0f : h;
    }
}

extern "C" void kernel_launch(void* const* d_in, const int* in_sizes, int n_in,
                              void* d_out, int out_size, void* d_ws, size_t ws_size,
                              hipStream_t stream) {
    const float* x_seq  = (const float*)d_in[0];   // (T, B, D) fp32
    const float* v_init = (const float*)d_in[1];   // (B, D)    fp32
    float*       out    = (float*)d_out;           // (T, B, D) fp32

    const int N = in_sizes[1];                 // B * D elements per timestep
    const int T = (N > 0) ? (in_sizes[0] / N) : 0;
    if (N <= 0 || T <= 0) return;

    const int n4 = N >> 2;                     // float4 lanes
    if (n4 > 0) {
        const int threads = 256;               // 8 wave32 waves per block
        const int blocks  = (n4 + threads - 1) / threads;
        if_scan_v4<<<blocks, threads, 0, stream>>>(x_seq, v_init, out, n4, T, N);
    }

    const int rem = N - (n4 << 2);
    if (rem > 0) {
        if_scan_tail<<<1, 256, 0, stream>>>(x_seq, v_init, out, n4 << 2, N, T, N);
    }
}